// NetworkWrapper_39290360824686
// MI455X (gfx1250) — compile-verified
//
#include <hip/hip_runtime.h>
#include <math.h>

// ---------------------------------------------------------------------------
// Problem constants (from the reference)
// ---------------------------------------------------------------------------
#define BB   8
#define CC   18
#define HH   256
#define WW   256
#define HWSZ (HH * WW)          // 65536
#define NHYP 1024
#define KK   9
#define NCOL (BB * KK)          // 72 softmax columns per side
#define LOG2INV 1.4426950408889634f

// Workspace layout (floats):
//  side s base = s*12 : [0]=vote_sum [1]=w_sum [2]=nll_sum [3]=kp_sum [4..11]=ent[b]
//  [24..95]  = yL per column (72)
//  [96..167] = yR per column (72)
#define WS_FLOATS 168
#define WS_YL 24
#define WS_YR 96

typedef __attribute__((ext_vector_type(2))) float v2f;
typedef __attribute__((ext_vector_type(8))) float v8f;

// ---------------------------------------------------------------------------
// Wave-wide f32 sum via V_WMMA_F32_16X16X4_F32 with an all-ones B matrix.
// Each lane contributes 2 values -> A is the 16x4 f32 matrix (64 vals/wave).
// D[m][n] = rowsum(m).  C/D layout: lane l<16 holds rows 0..7 in its 8 VGPRs,
// lane l>=16 holds rows 8..15, so sum(c[0..7]) + shfl_xor(.,16) = full sum.
// Requires EXEC == all ones (call sites are in uniform control flow with
// exact-fit launch geometry).
// ---------------------------------------------------------------------------
__device__ __forceinline__ float wave_sum2(float x0, float x1) {
    v2f a; a[0] = x0; a[1] = x1;
    v2f b; b[0] = 1.0f; b[1] = 1.0f;
    v8f c = {};
    c = __builtin_amdgcn_wmma_f32_16x16x4_f32(false, a, false, b,
                                              (short)0, c, false, false);
    float s = c[0] + c[1] + c[2] + c[3] + c[4] + c[5] + c[6] + c[7];
    s += __shfl_xor(s, 16, 32);
    return s; // full wave sum, on every lane
}

// Block-wide sum (8 waves / 256 threads); returns total to all threads.
__device__ __forceinline__ float block_sum2(float x0, float x1, float* sh) {
    const int lane = threadIdx.x & 31;
    const int wave = threadIdx.x >> 5;
    float ws = wave_sum2(x0, x1);
    __syncthreads();                 // protect sh reuse across calls
    if (lane == 0) sh[wave] = ws;
    __syncthreads();
    float tot = 0.f;
#pragma unroll
    for (int i = 0; i < 8; ++i) tot += sh[i];
    return tot;
}

// Block-wide max (8 waves / 256 threads); returns max to all threads.
__device__ __forceinline__ float block_max(float x, float* sh) {
#pragma unroll
    for (int off = 16; off > 0; off >>= 1)
        x = fmaxf(x, __shfl_xor(x, off, 32));
    const int lane = threadIdx.x & 31;
    const int wave = threadIdx.x >> 5;
    __syncthreads();
    if (lane == 0) sh[wave] = x;
    __syncthreads();
    float m = sh[0];
#pragma unroll
    for (int i = 1; i < 8; ++i) m = fmaxf(m, sh[i]);
    return m;
}

__device__ __forceinline__ unsigned hash32(unsigned x) {
    x ^= x >> 16; x *= 0x7feb352dU;
    x ^= x >> 15; x *= 0x846ca68bU;
    x ^= x >> 16;
    return x;
}

// ---------------------------------------------------------------------------
// Kernel 0: zero the workspace accumulators
// ---------------------------------------------------------------------------
__global__ void zero_kernel(float* __restrict__ ws) {
    int i = threadIdx.x;
    if (i < WS_FLOATS) ws[i] = 0.f;
}

// ---------------------------------------------------------------------------
// Kernel 1: streaming vote (smooth-L1) + mask count + seg NLL for one side.
// 512 blocks x 256 threads = 131072 threads; each handles exactly one group
// of 4 consecutive pixels (b128 loads) -> EXEC stays full for the WMMA reduce.
// ---------------------------------------------------------------------------
__global__ void vote_seg_kernel(const float* __restrict__ pred,
                                const float* __restrict__ gt,
                                const int*   __restrict__ mask,
                                const float* __restrict__ seg,
                                float* __restrict__ acc /* [vote,wsum,nll] */) {
    const int P4 = (BB * HWSZ) / 4;                 // 131072 pixel-quads
    const int tid = blockIdx.x * blockDim.x + threadIdx.x;
    const int stride = gridDim.x * blockDim.x;

    float v0 = 0.f, v1 = 0.f;                       // vote partial pair
    float w0 = 0.f, w1 = 0.f;                       // mask-count pair
    float n0 = 0.f, n1 = 0.f;                       // NLL pair

    for (int g = tid; g < P4; g += stride) {        // exact fit: 1 iteration
        const int p  = g << 2;
        const int b  = p >> 16;                     // / HWSZ
        const int hw = p & 0xFFFF;                  // % HWSZ

        const int4 m4 = *(const int4*)(mask + p);
        float w[4] = {(float)m4.x, (float)m4.y, (float)m4.z, (float)m4.w};
        w0 += w[0] + w[2];
        w1 += w[1] + w[3];

        // seg: 2-class log-softmax NLL at label = mask
        const float4 s0 = *(const float4*)(seg + ((size_t)(b * 2 + 0)) * HWSZ + hw);
        const float4 s1 = *(const float4*)(seg + ((size_t)(b * 2 + 1)) * HWSZ + hw);
        {
            float a0[4] = {s0.x, s0.y, s0.z, s0.w};
            float a1[4] = {s1.x, s1.y, s1.z, s1.w};
#pragma unroll
            for (int j = 0; j < 4; ++j) {
                float mx  = fmaxf(a0[j], a1[j]);
                float lse = mx + logf(expf(a0[j] - mx) + expf(a1[j] - mx));
                float nll = lse - ((w[j] != 0.f) ? a1[j] : a0[j]);
                if (j & 1) n1 += nll; else n0 += nll;
            }
        }

        // vote: smooth-L1( pred*w - gt*w ) summed over 18 channels
        const float* pb = pred + (size_t)b * CC * HWSZ + hw;
        const float* gb = gt   + (size_t)b * CC * HWSZ + hw;
#pragma unroll 3
        for (int c = 0; c < CC; ++c) {
            if (c + 2 < CC) {                        // uniform guard
                __builtin_prefetch(pb + (size_t)(c + 2) * HWSZ, 0, 1);
                __builtin_prefetch(gb + (size_t)(c + 2) * HWSZ, 0, 1);
            }
            const float4 pv = *(const float4*)(pb + (size_t)c * HWSZ);
            const float4 gv = *(const float4*)(gb + (size_t)c * HWSZ);
            float pa[4] = {pv.x, pv.y, pv.z, pv.w};
            float ga[4] = {gv.x, gv.y, gv.z, gv.w};
#pragma unroll
            for (int j = 0; j < 4; ++j) {
                float d = fabsf(pa[j] - ga[j]) * w[j];
                float f = (d < 1.f) ? 0.5f * d * d : d - 0.5f;
                if ((c + j) & 1) v1 += f; else v0 += f;
            }
        }
    }

    // wave reduction on the matrix pipe; EXEC is all-ones here
    float vs = wave_sum2(v0, v1);
    float ws = wave_sum2(w0, w1);
    float ns = wave_sum2(n0, n1);
    if ((threadIdx.x & 31) == 0) {
        atomicAdd(&acc[0], vs);
        atomicAdd(&acc[1], ws);
        atomicAdd(&acc[2], ns);
    }
}

// ---------------------------------------------------------------------------
// Kernel 2: per-column (b,k) softmax over 1024 hypotheses:
//   expected keypoint distance, entropy (bits), Gumbel-argmax sample y-coord.
// 72 blocks x 256 threads; each thread owns 4 hypotheses.
// ---------------------------------------------------------------------------
__global__ void kp_kernel(const float* __restrict__ keypoints,
                          const float* __restrict__ scores,
                          const float* __restrict__ kpt2d,
                          const float* __restrict__ offset,
                          const float* __restrict__ a_ptr,
                          float* __restrict__ kp_acc,   // [1]
                          float* __restrict__ ent_acc,  // [BB]
                          float* __restrict__ y_out,    // [NCOL]
                          unsigned seed) {
    __shared__ float sh[8];
    __shared__ float shz[8];
    __shared__ int   shn[8];

    const int col = blockIdx.x;
    const int b = col / KK, k = col % KK;
    const int t = threadIdx.x;
    const float a = a_ptr[0];
    const float tx = kpt2d[(b * KK + k) * 2 + 0];
    const float ty = kpt2d[(b * KK + k) * 2 + 1];

    float s[4], dist[4];
    int   n[4];
#pragma unroll
    for (int i = 0; i < 4; ++i) {
        n[i] = t + i * 256;
        size_t idx = ((size_t)(b * NHYP + n[i])) * KK + k;
        s[i] = a * scores[idx];
        float dx = keypoints[idx * 2 + 0] - tx;
        float dy = keypoints[idx * 2 + 1] - ty;
        dist[i] = sqrtf(dx * dx + dy * dy);
    }

    float mx = block_max(fmaxf(fmaxf(s[0], s[1]), fmaxf(s[2], s[3])), sh);

    float e[4];
#pragma unroll
    for (int i = 0; i < 4; ++i) e[i] = expf(s[i] - mx);
    const float S = block_sum2(e[0] + e[2], e[1] + e[3], sh);
    const float invS = 1.f / S;
    const float lS = logf(S);

    float kp0 = 0.f, kp1 = 0.f, en0 = 0.f, en1 = 0.f;
#pragma unroll
    for (int i = 0; i < 4; ++i) {
        float pr = e[i] * invS;
        float lp = (s[i] - mx - lS) * LOG2INV;   // log-softmax in bits
        if (i & 1) { kp1 += pr * dist[i]; en1 += pr * lp; }
        else       { kp0 += pr * dist[i]; en0 += pr * lp; }
    }
    const float kp  = block_sum2(kp0, kp1, sh);
    const float ent = -block_sum2(en0, en1, sh);

    // Gumbel-argmax categorical sample (deterministic counter hash)
    float bz = -1e30f; int bn = 0;
#pragma unroll
    for (int i = 0; i < 4; ++i) {
        unsigned h = hash32(seed + 0x9e3779b9u * (unsigned)(col * NHYP + n[i]));
        float u = ((float)(h >> 8) + 0.5f) * (1.0f / 16777216.0f); // (0,1)
        float z = s[i] - logf(-logf(u));
        if (z > bz || (z == bz && n[i] < bn)) { bz = z; bn = n[i]; }
    }
#pragma unroll
    for (int off = 16; off > 0; off >>= 1) {
        float oz = __shfl_xor(bz, off, 32);
        int   on = __shfl_xor(bn, off, 32);
        if (oz > bz || (oz == bz && on < bn)) { bz = oz; bn = on; }
    }
    __syncthreads();
    if ((t & 31) == 0) { shz[t >> 5] = bz; shn[t >> 5] = bn; }
    __syncthreads();
    if (t == 0) {
        float z = shz[0]; int widx = shn[0];
        for (int i = 1; i < 8; ++i)
            if (shz[i] > z || (shz[i] == z && shn[i] < widx)) { z = shz[i]; widx = shn[i]; }
        size_t yi = (((size_t)(b * NHYP + widx)) * KK + k) * 2 + 1;
        y_out[col] = keypoints[yi] + offset[b * 2 + 1];
        atomicAdd(&kp_acc[0], kp);
        atomicAdd(&ent_acc[b], ent / (float)KK);
    }
}

// ---------------------------------------------------------------------------
// Kernel 3: scalar epilogue combining everything into the final loss.
// ---------------------------------------------------------------------------
__global__ void final_kernel(const float* __restrict__ ws,
                             const int* __restrict__ epoch_p,
                             float* __restrict__ out) {
    if (threadIdx.x != 0 || blockIdx.x != 0) return;

    float vote[2], segl[2], kpl[2], entl[2];
#pragma unroll
    for (int sd = 0; sd < 2; ++sd) {
        const float* w = ws + sd * 12;
        float vs = w[0], wsum = w[1], nll = w[2], kps = w[3];
        vote[sd] = (wsum > 0.f) ? vs / fmaxf(wsum, 1.f) / (float)CC : vs;
        segl[sd] = nll / (float)(BB * HWSZ);
        kpl[sd]  = kps / (float)NCOL;
        float e = 0.f;
        for (int b = 0; b < BB; ++b) e += fabsf(w[4 + b] - 6.0f);
        entl[sd] = e / (float)BB;
    }
    float vote_loss = 0.5f * (vote[0] + vote[1]);
    float seg_loss  = 0.5f * (segl[0] + segl[1]);
    float kp_loss   = 0.5f * (kpl[0]  + kpl[1]);
    float ent_loss  = 0.5f * (entl[0] + entl[1]);

    float kp_w   = 1.f / (1.f + expf(-0.5f * (20.0f - kp_loss)));
    float vote_w = 1.f - kp_w;

    float vertex;
    if (epoch_p[0] > 49) {
        float epi = 0.f;
        for (int i = 0; i < NCOL; ++i) {
            float d = fabsf(ws[WS_YL + i] - ws[WS_YR + i]);
            if (!isfinite(d)) d = 0.f;
            epi += d;
        }
        epi /= (float)NCOL;
        vertex = kp_w * (kp_loss + epi) + vote_w * vote_loss;
    } else {
        vertex = kp_w * kp_loss + vote_w * vote_loss;
    }
    out[0] = vertex + ent_loss + seg_loss;
}

// ---------------------------------------------------------------------------
// Host launcher
// ---------------------------------------------------------------------------
extern "C" void kernel_launch(void* const* d_in, const int* in_sizes, int n_in,
                              void* d_out, int out_size, void* d_ws, size_t ws_size,
                              hipStream_t stream) {
    (void)in_sizes; (void)n_in; (void)out_size; (void)ws_size;

    const float* vpL  = (const float*)d_in[0];
    const float* gtL  = (const float*)d_in[1];
    const int*   mL   = (const int*)  d_in[2];
    const float* segL = (const float*)d_in[3];
    const float* kpsL = (const float*)d_in[4];
    const float* scL  = (const float*)d_in[5];
    const float* k2L  = (const float*)d_in[6];
    const float* ofL  = (const float*)d_in[7];
    const float* vpR  = (const float*)d_in[8];
    const float* gtR  = (const float*)d_in[9];
    const int*   mR   = (const int*)  d_in[10];
    const float* segR = (const float*)d_in[11];
    const float* kpsR = (const float*)d_in[12];
    const float* scR  = (const float*)d_in[13];
    const float* k2R  = (const float*)d_in[14];
    const float* ofR  = (const float*)d_in[15];
    const float* a    = (const float*)d_in[16];
    const int*   ep   = (const int*)  d_in[17];

    float* ws  = (float*)d_ws;
    float* out = (float*)d_out;

    zero_kernel<<<1, 256, 0, stream>>>(ws);

    // 512 blocks x 256 threads == exactly (B*H*W)/4 pixel-quads -> EXEC full
    vote_seg_kernel<<<512, 256, 0, stream>>>(vpL, gtL, mL, segL, ws + 0);
    vote_seg_kernel<<<512, 256, 0, stream>>>(vpR, gtR, mR, segR, ws + 12);

    kp_kernel<<<NCOL, 256, 0, stream>>>(kpsL, scL, k2L, ofL, a,
                                        ws + 3, ws + 4, ws + WS_YL, 0x1234u);
    kp_kernel<<<NCOL, 256, 0, stream>>>(kpsR, scR, k2R, ofR, a,
                                        ws + 15, ws + 16, ws + WS_YR, 0x5678u);

    final_kernel<<<1, 32, 0, stream>>>(ws, ep, out);
}